// MultiLinkAcrobot_23235773071848
// MI455X (gfx1250) — compile-verified
//
#include <hip/hip_runtime.h>
#include <hip/hip_bf16.h>

typedef float v2f __attribute__((ext_vector_type(2)));
typedef float v8f __attribute__((ext_vector_type(8)));

#define QD 8

__device__ __forceinline__ constexpr int TIX(int i, int j) { return i * (i - 1) / 2 + j; }

// V_WMMA_F32_16X16X4_F32: D(16x16 f32) = A(16x4 f32) * B(4x16 f32) + C
__device__ __forceinline__ v8f wmma_f32_16x16x4(v2f a, v2f b, v8f c) {
  return __builtin_amdgcn_wmma_f32_16x16x4_f32(
      /*neg_a=*/false, a, /*neg_b=*/false, b,
      /*c_mod=*/(short)0, c, /*reuse_a=*/false, /*reuse_b=*/false);
}

// ---------------------------------------------------------------------------
// Setup: derive per-link uniform coefficients from params (25 floats).
// Layout in d_ws (floats):
//   [0..27]   rl[i>j]   = ls_i*ls_j*(ms_suf_i - 0.5*ms_i)   (i*(i-1)/2 + j)
//   [28..35]  Is_suf
//   [36..43]  D (mass-matrix diagonal)
//   [44..51]  gcoef_i   = g*ls_i*(ms_suf_i - 0.5*ms_i)
//   [52..59]  taus
// ---------------------------------------------------------------------------
__global__ void setup_coeffs_kernel(const float* __restrict__ p, float* __restrict__ w) {
  if (threadIdx.x != 0 || blockIdx.x != 0) return;
  float ms[QD], ls[QD], Is[QD];
  for (int i = 0; i < QD; ++i) {
    ms[i] = p[i];
    ls[i] = p[QD + i];
    Is[i] = ms[i] * ls[i] * ls[i] * (1.0f / 12.0f);
  }
  float g = p[2 * QD];
  float ms_suf[QD], Is_suf[QD];
  float am = 0.0f, ai = 0.0f;
  for (int i = QD - 1; i >= 0; --i) {
    am += ms[i]; ai += Is[i];
    ms_suf[i] = am; Is_suf[i] = ai;
  }
  for (int i = 1; i < QD; ++i) {
    float ks = ms_suf[i] - 0.5f * ms[i];
    for (int j = 0; j < i; ++j) w[TIX(i, j)] = ls[i] * ls[j] * ks;
  }
  for (int i = 0; i < QD; ++i) {
    w[28 + i] = Is_suf[i];
    w[36 + i] = ls[i] * ls[i] * (ms_suf[i] - 0.75f * ms[i]) + Is_suf[i];
    w[44 + i] = g * ls[i] * (ms_suf[i] - 0.5f * ms[i]);
    w[52 + i] = p[2 * QD + 1 + i];
  }
}

// ---------------------------------------------------------------------------
// Main kernel: one sample per lane; prefix sums (cs = L*q, P = L*v) done with
// V_WMMA_F32_16X16X4_F32 chains (shared triangular operand), everything else
// scalar f32 in registers; unrolled Cholesky solve.
// ---------------------------------------------------------------------------
__global__ void __launch_bounds__(256) acrobot_dyn_kernel(
    const float* __restrict__ gq, const float* __restrict__ gv,
    const float* __restrict__ gu, const float* __restrict__ cw,
    float* __restrict__ gout) {
  __shared__ float smem[4096];  // 16 KB: staging (q:[0,2048), v:[2048,4096)) then results [0,4096)

  const int tid  = threadIdx.x;
  const int gid  = blockIdx.x * 256 + tid;
  const int lane = tid & 31;
  const int wave = tid >> 5;
  const int n16  = lane & 15;   // N column (B/C layout) / M row (A layout)
  const int half = lane >> 4;   // 0: lanes 0-15, 1: lanes 16-31

  // ---- load q, v (8 f32 each, contiguous per sample) ----
  const float4* q4 = (const float4*)gq;
  const float4* v4 = (const float4*)gv;
  float4 qa = q4[gid * 2 + 0], qb = q4[gid * 2 + 1];
  float4 va = v4[gid * 2 + 0], vb = v4[gid * 2 + 1];
  float vv[QD] = {va.x, va.y, va.z, va.w, vb.x, vb.y, vb.z, vb.w};

  ((float4*)smem)[tid * 2 + 0] = qa;
  ((float4*)smem)[tid * 2 + 1] = qb;
  ((float4*)(smem + 2048))[tid * 2 + 0] = va;
  ((float4*)(smem + 2048))[tid * 2 + 1] = vb;
  __syncthreads();

  // ---- B fragments: lower-triangular-ones (prefix sum), built per-lane.
  // B 4x16 layout: VGPR0 = row K(=0|2 + 4*cb), VGPR1 = row K(=1|3 + 4*cb); N = lane&15.
  // q-path fills C columns 0..7, v-path fills C columns 8..15.
  v2f bq[2], bv[2];
#pragma unroll
  for (int cb = 0; cb < 2; ++cb) {
    int k0 = 4 * cb + 2 * half;
    bq[cb][0] = (n16 < 8 && k0     <= n16)     ? 1.0f : 0.0f;
    bq[cb][1] = (n16 < 8 && k0 + 1 <= n16)     ? 1.0f : 0.0f;
    bv[cb][0] = (n16 >= 8 && k0     <= n16 - 8) ? 1.0f : 0.0f;
    bv[cb][1] = (n16 >= 8 && k0 + 1 <= n16 - 8) ? 1.0f : 0.0f;
  }

  // ---- 2 tiles of 16 samples per wave; 4 chained WMMAs each (K=8 q + K=8 v) ----
  v8f acc[2];
#pragma unroll
  for (int t = 0; t < 2; ++t) {
    int row = wave * 32 + t * 16 + n16;  // sample-in-block = A-matrix row M
    v8f c = {0.f, 0.f, 0.f, 0.f, 0.f, 0.f, 0.f, 0.f};
#pragma unroll
    for (int cb = 0; cb < 2; ++cb) {     // A 16x4 layout: VGPR0=K(0|2), VGPR1=K(1|3)
      v2f aq = *(const v2f*)(smem + row * 8 + 4 * cb + 2 * half);
      c = wmma_f32_16x16x4(aq, bq[cb], c);
    }
#pragma unroll
    for (int cb = 0; cb < 2; ++cb) {
      v2f av = *(const v2f*)(smem + 2048 + row * 8 + 4 * cb + 2 * half);
      c = wmma_f32_16x16x4(av, bv[cb], c);
    }
    acc[t] = c;
  }
  __syncthreads();

  // ---- scatter C back (C layout: VGPR r -> M = r + 8*half, N = lane&15) ----
#pragma unroll
  for (int t = 0; t < 2; ++t)
#pragma unroll
    for (int r = 0; r < 8; ++r)
      smem[(wave * 32 + t * 16 + r + 8 * half) * 16 + n16] = acc[t][r];
  __syncthreads();

  float4 c0 = ((const float4*)smem)[tid * 4 + 0];
  float4 c1 = ((const float4*)smem)[tid * 4 + 1];
  float4 p0 = ((const float4*)smem)[tid * 4 + 2];
  float4 p1 = ((const float4*)smem)[tid * 4 + 3];
  float cs[QD] = {c0.x, c0.y, c0.z, c0.w, c1.x, c1.y, c1.z, c1.w};  // cumsum(q)
  float P[QD]  = {p0.x, p0.y, p0.z, p0.w, p1.x, p1.y, p1.z, p1.w};  // cumsum(v)

  // ---- 8 sincos; all pair angles via difference identities ----
  float s_[QD], c_[QD];
#pragma unroll
  for (int k = 0; k < QD; ++k) __sincosf(cs[k], &s_[k], &c_[k]);

  float vp[QD];
#pragma unroll
  for (int k = 0; k < QD; ++k) vp[k] = vv[k] * P[k];

  float W[QD], Z[QD], Ml[28];
#pragma unroll
  for (int k = 0; k < QD; ++k) { W[k] = 0.0f; Z[k] = 0.0f; }

#pragma unroll
  for (int i = 1; i < QD; ++i) {
#pragma unroll
    for (int j = 0; j < i; ++j) {
      const int px = TIX(i, j);
      float cd = c_[i] * c_[j] + s_[i] * s_[j];   // cos(cs_i - cs_j)
      float sd = s_[i] * c_[j] - c_[i] * s_[j];   // sin(cs_i - cs_j)
      float rl = cw[px];                          // ls_i*ls_j*(ms_suf_i - 0.5 ms_i)
      Ml[px] = fmaf(rl, cd, cw[28 + i]);          // M_ij = A_ij*cd + Is_suf[max]
      float e = rl * sd;                          // antisymmetric Coriolis kernel
      W[i] = fmaf(e, vv[j], W[i]);  W[j] = fmaf(-e, vv[i], W[j]);
      Z[i] = fmaf(e, vp[j], Z[i]);  Z[j] = fmaf(-e, vp[i], Z[j]);
    }
  }

  const float4* u4 = (const float4*)gu;
  float4 ua = u4[gid * 2 + 0], ub = u4[gid * 2 + 1];
  float uu[QD] = {ua.x, ua.y, ua.z, ua.w, ub.x, ub.y, ub.z, ub.w};

  // rhs = taus*u - Cv - G ;  Cv_k = -P_k W_k + Z_k + suffix(v*W)_k ; G_k = suffix(gcoef*s)_k
  float rhs[QD];
  float ys = 0.0f, gs = 0.0f;
#pragma unroll
  for (int k = QD - 1; k >= 0; --k) {
    ys = fmaf(vv[k], W[k], ys);
    gs = fmaf(cw[44 + k], s_[k], gs);
    rhs[k] = fmaf(cw[52 + k], uu[k], fmaf(P[k], W[k], -Z[k]) - ys - gs);
  }

  // ---- Cholesky LL^T on SPD M (lower-packed in Ml, diag from cw) ----
  float il[QD];
#pragma unroll
  for (int j = 0; j < QD; ++j) {
    float d = cw[36 + j];
#pragma unroll
    for (int k = 0; k < j; ++k) d = fmaf(-Ml[TIX(j, k)], Ml[TIX(j, k)], d);
    float sq = sqrtf(d);
    il[j] = 1.0f / sq;
#pragma unroll
    for (int i = j + 1; i < QD; ++i) {
      float sum = Ml[TIX(i, j)];
#pragma unroll
      for (int k = 0; k < j; ++k) sum = fmaf(-Ml[TIX(i, k)], Ml[TIX(j, k)], sum);
      Ml[TIX(i, j)] = sum * il[j];
    }
  }
  float x[QD];
#pragma unroll
  for (int i = 0; i < QD; ++i) {          // L y = rhs
    float sum = rhs[i];
#pragma unroll
    for (int k = 0; k < i; ++k) sum = fmaf(-Ml[TIX(i, k)], x[k], sum);
    x[i] = sum * il[i];
  }
#pragma unroll
  for (int i = QD - 1; i >= 0; --i) {     // L^T x = y
    float sum = x[i];
#pragma unroll
    for (int k = i + 1; k < QD; ++k) sum = fmaf(-Ml[TIX(k, i)], x[k], sum);
    x[i] = sum * il[i];
  }

  ((float4*)gout)[gid * 2 + 0] = make_float4(x[0], x[1], x[2], x[3]);
  ((float4*)gout)[gid * 2 + 1] = make_float4(x[4], x[5], x[6], x[7]);
}

extern "C" void kernel_launch(void* const* d_in, const int* in_sizes, int n_in,
                              void* d_out, int out_size, void* d_ws, size_t ws_size,
                              hipStream_t stream) {
  (void)n_in; (void)out_size; (void)ws_size;
  const float* q = (const float*)d_in[0];
  const float* v = (const float*)d_in[1];
  const float* u = (const float*)d_in[2];
  const float* p = (const float*)d_in[3];
  float* cw  = (float*)d_ws;
  float* out = (float*)d_out;

  const int nB = in_sizes[0] / QD;        // 131072 samples
  setup_coeffs_kernel<<<1, 32, 0, stream>>>(p, cw);
  acrobot_dyn_kernel<<<nB / 256, 256, 0, stream>>>(q, v, u, cw, out);
}